// GAT_model_45896020525325
// MI455X (gfx1250) — compile-verified
//
#include <hip/hip_runtime.h>
#include <hip/hip_bf16.h>

typedef __attribute__((ext_vector_type(2))) float v2f;
typedef __attribute__((ext_vector_type(8))) float v8f;

#define HDIM 512
#define TILE_M 32
#define TILE_N 64
#define TILE_K 32

// ---------------------------------------------------------------------------
// GEMM: C[M x Nout] = A[M x K] @ B[K x Nout] + bias (optional ReLU)
// Requires K % 32 == 0, Nout % 64 == 0. One 16x16 WMMA tile per wave.
// ---------------------------------------------------------------------------
__global__ __launch_bounds__(256) void gemm_wmma_f32(
    const float* __restrict__ A, const float* __restrict__ B,
    const float* __restrict__ bias, float* __restrict__ C,
    int M, int K, int Nout, int relu)
{
    __shared__ float As[TILE_M][TILE_K + 1];
    __shared__ float Bs[TILE_K][TILE_N + 1];

    const int tid  = threadIdx.x;
    const int wave = tid >> 5;          // 0..7
    const int lane = tid & 31;
    const int half = lane >> 4;         // 0 or 1
    const int l    = lane & 15;

    const int block_m = blockIdx.y * TILE_M;
    const int block_n = blockIdx.x * TILE_N;
    const int wm = (wave >> 2) * 16;    // 0 / 16
    const int wn = (wave & 3) * 16;     // 0,16,32,48

    v8f acc = {};

    for (int k0 = 0; k0 < K; k0 += TILE_K) {
        // stage A: 32x32
        for (int i = tid; i < TILE_M * TILE_K; i += 256) {
            int r = i >> 5, c = i & 31;
            int gr = block_m + r;
            As[r][c] = (gr < M) ? A[(size_t)gr * K + k0 + c] : 0.0f;
        }
        // stage B: 32x64
        for (int i = tid; i < TILE_K * TILE_N; i += 256) {
            int r = i >> 6, c = i & 63;
            Bs[r][c] = B[(size_t)(k0 + r) * Nout + block_n + c];
        }
        __syncthreads();

        #pragma unroll
        for (int kk = 0; kk < TILE_K; kk += 4) {
            v2f a, b;
            // A 16x4 frag: lanes0-15 -> K=kk,kk+1 ; lanes16-31 -> K=kk+2,kk+3
            a.x = As[wm + l][kk + 2 * half];
            a.y = As[wm + l][kk + 2 * half + 1];
            // B 4x16 frag: VGPR0 rows {kk, kk+2}, VGPR1 rows {kk+1, kk+3}
            b.x = Bs[kk + 2 * half][wn + l];
            b.y = Bs[kk + 2 * half + 1][wn + l];
            acc = __builtin_amdgcn_wmma_f32_16x16x4_f32(
                false, a, false, b, (short)0, acc, false, false);
        }
        __syncthreads();
    }

    const int col = block_n + wn + l;
    const float bv = bias ? bias[col] : 0.0f;
    #pragma unroll
    for (int r = 0; r < 8; ++r) {
        int row = block_m + wm + r + 8 * half;   // C/D: VGPR r -> M=r / r+8
        if (row < M) {
            float v = acc[r] + bv;
            if (relu) v = v > 0.0f ? v : 0.0f;
            C[(size_t)row * Nout + col] = v;
        }
    }
}

// ---------------------------------------------------------------------------
// Zero the scatter target, init segment max (-inf) and segment sum (0)
// ---------------------------------------------------------------------------
__global__ void init_layer(float* __restrict__ out, float* __restrict__ m,
                           float* __restrict__ s, int N)
{
    int i = blockIdx.x * blockDim.x + threadIdx.x;
    if (i < N * HDIM) out[i] = 0.0f;
    if (i < N) { m[i] = -__builtin_inff(); s[i] = 0.0f; }
}

// ---------------------------------------------------------------------------
// Per-edge attention logit: e = att . leaky_relu(xl[src] + xr[dst], 0.2)
// One wave32 per edge; lane0 does segment-max via float atomicMax.
// Edges i >= E are self-loops (src = dst = i - E).
// ---------------------------------------------------------------------------
__global__ __launch_bounds__(256) void edge_score(
    const long long* __restrict__ ei, int E, int N,
    const float* __restrict__ xl, const float* __restrict__ xr,
    const float* __restrict__ att,
    float* __restrict__ escore, float* __restrict__ m)
{
    int wg   = blockIdx.x * (blockDim.x >> 5) + (threadIdx.x >> 5);
    int lane = threadIdx.x & 31;
    int E2 = E + N;
    if (wg >= E2) return;
    int s, d;
    if (wg < E) { s = (int)ei[wg]; d = (int)ei[E + wg]; }
    else        { s = d = wg - E; }

    const float* pl = xl + (size_t)s * HDIM;
    const float* pr = xr + (size_t)d * HDIM;
    float acc = 0.0f;
    #pragma unroll 4
    for (int h = lane; h < HDIM; h += 32) {
        float z = pl[h] + pr[h];
        z = z > 0.0f ? z : 0.2f * z;
        acc += z * att[h];
    }
    #pragma unroll
    for (int off = 16; off > 0; off >>= 1) acc += __shfl_down(acc, off, 32);
    if (lane == 0) {
        escore[wg] = acc;
        atomicMax(&m[d], acc);
    }
}

// ---------------------------------------------------------------------------
// p = exp(e - m[dst]); segment sum via atomicAdd (overwrites e with p)
// ---------------------------------------------------------------------------
__global__ void edge_exp(const long long* __restrict__ ei, int E, int N,
                         float* __restrict__ escore,
                         const float* __restrict__ m, float* __restrict__ s)
{
    int i = blockIdx.x * blockDim.x + threadIdx.x;
    int E2 = E + N;
    if (i >= E2) return;
    int d = (i < E) ? (int)ei[E + i] : (i - E);
    float p = expf(escore[i] - m[d]);
    escore[i] = p;
    atomicAdd(&s[d], p);
}

// ---------------------------------------------------------------------------
// out[dst] += alpha * xl[src], alpha = p / (s[dst] + 1e-16). Wave per edge.
// ---------------------------------------------------------------------------
__global__ __launch_bounds__(256) void edge_scatter(
    const long long* __restrict__ ei, int E, int N,
    const float* __restrict__ p, const float* __restrict__ ssum,
    const float* __restrict__ xl, float* __restrict__ out)
{
    int wg   = blockIdx.x * (blockDim.x >> 5) + (threadIdx.x >> 5);
    int lane = threadIdx.x & 31;
    int E2 = E + N;
    if (wg >= E2) return;
    int s, d;
    if (wg < E) { s = (int)ei[wg]; d = (int)ei[E + wg]; }
    else        { s = d = wg - E; }
    float alpha = p[wg] / (ssum[d] + 1e-16f);
    const float* pl = xl + (size_t)s * HDIM;
    float* po = out + (size_t)d * HDIM;
    #pragma unroll 4
    for (int h = lane; h < HDIM; h += 32) atomicAdd(&po[h], alpha * pl[h]);
}

// ---------------------------------------------------------------------------
// h = relu(out + b)  (in-place ok)
// ---------------------------------------------------------------------------
__global__ void bias_relu(const float* __restrict__ in,
                          const float* __restrict__ b,
                          float* __restrict__ out, int N)
{
    int i = blockIdx.x * blockDim.x + threadIdx.x;
    if (i >= N * HDIM) return;
    float v = in[i] + b[i & (HDIM - 1)];
    out[i] = v > 0.0f ? v : 0.0f;
}

// ---------------------------------------------------------------------------
// Final 256 -> 2 linear + softmax. d_out = [logits (N x 2) | probs (N x 2)]
// ---------------------------------------------------------------------------
__global__ void head_kernel(const float* __restrict__ hin,
                            const float* __restrict__ w,
                            const float* __restrict__ b,
                            float* __restrict__ out, int N)
{
    int i = blockIdx.x * blockDim.x + threadIdx.x;
    if (i >= N) return;
    const float* h = hin + (size_t)i * 256;
    float l0 = b[0], l1 = b[1];
    #pragma unroll 4
    for (int k = 0; k < 256; ++k) {
        float v = h[k];
        l0 += v * w[2 * k];
        l1 += v * w[2 * k + 1];
    }
    out[2 * i] = l0;
    out[2 * i + 1] = l1;
    float mx = fmaxf(l0, l1);
    float e0 = expf(l0 - mx), e1 = expf(l1 - mx);
    float inv = 1.0f / (e0 + e1);
    out[2 * N + 2 * i]     = e0 * inv;
    out[2 * N + 2 * i + 1] = e1 * inv;
}

// ---------------------------------------------------------------------------
extern "C" void kernel_launch(void* const* d_in, const int* in_sizes, int n_in,
                              void* d_out, int out_size, void* d_ws, size_t ws_size,
                              hipStream_t stream)
{
    const int DIN = 1024;
    const float* x = (const float*)d_in[0];
    const long long* ei = (const long long*)d_in[1];
    const int N  = in_sizes[0] / DIN;
    const int E  = in_sizes[1] / 2;
    const int E2 = E + N;

    // workspace layout (floats)
    float* B0   = (float*)d_ws;
    float* B1   = B0 + (size_t)N * HDIM;
    float* XL   = B1 + (size_t)N * HDIM;
    float* XR   = XL + (size_t)N * HDIM;
    float* EBUF = XR + (size_t)N * HDIM;   // E2 scores / probs
    float* MBUF = EBUF + E2;               // N segment max
    float* SBUF = MBUF + N;                // N segment sum

    const int waves_per_block = 8;
    const dim3 blk(256);
    const int gemm_rows = (N + TILE_M - 1) / TILE_M;
    const int grid_elems  = (N * HDIM + 255) / 256;
    const int grid_edges  = (E2 + 255) / 256;
    const int grid_edgeswave = (E2 + waves_per_block - 1) / waves_per_block;

    const float* hin = x;
    int K = DIN;
    float* hout = B0;

    for (int layer = 0; layer < 4; ++layer) {
        const float* wl   = (const float*)d_in[2 + 6 * layer];
        const float* bl   = (const float*)d_in[3 + 6 * layer];
        const float* wr   = (const float*)d_in[4 + 6 * layer];
        const float* br   = (const float*)d_in[5 + 6 * layer];
        const float* att  = (const float*)d_in[6 + 6 * layer];
        const float* bvec = (const float*)d_in[7 + 6 * layer];

        dim3 ggrid(HDIM / TILE_N, gemm_rows);
        gemm_wmma_f32<<<ggrid, blk, 0, stream>>>(hin, wl, bl, XL, N, K, HDIM, 0);
        gemm_wmma_f32<<<ggrid, blk, 0, stream>>>(hin, wr, br, XR, N, K, HDIM, 0);

        init_layer<<<grid_elems, blk, 0, stream>>>(hout, MBUF, SBUF, N);
        edge_score<<<grid_edgeswave, blk, 0, stream>>>(ei, E, N, XL, XR, att, EBUF, MBUF);
        edge_exp<<<grid_edges, blk, 0, stream>>>(ei, E, N, EBUF, MBUF, SBUF);
        edge_scatter<<<grid_edgeswave, blk, 0, stream>>>(ei, E, N, EBUF, SBUF, XL, hout);
        bias_relu<<<grid_elems, blk, 0, stream>>>(hout, bvec, hout, N);

        hin = hout;
        K = HDIM;
        hout = (hout == B0) ? B1 : B0;
    }
    // hin = layer-4 output (B1), hout = B0 free

    const float* lw1 = (const float*)d_in[26];
    const float* lb1 = (const float*)d_in[27];
    const float* lw2 = (const float*)d_in[28];
    const float* lb2 = (const float*)d_in[29];
    const float* lw3 = (const float*)d_in[30];
    const float* lb3 = (const float*)d_in[31];

    dim3 g1(512 / TILE_N, gemm_rows);
    gemm_wmma_f32<<<g1, blk, 0, stream>>>(hin, lw1, lb1, hout, N, 512, 512, 1);
    dim3 g2(256 / TILE_N, gemm_rows);
    gemm_wmma_f32<<<g2, blk, 0, stream>>>(hout, lw2, lb2, XL, N, 512, 256, 1);
    head_kernel<<<(N + 255) / 256, blk, 0, stream>>>(XL, lw3, lb3, (float*)d_out, N);
}